// GCN_72988674228318
// MI455X (gfx1250) — compile-verified
//
#include <hip/hip_runtime.h>

#define N_NODES 50000
#define IN_C    128
#define HID_C   128
#define OUT_C   64
#define N_EDGES 600000

typedef __attribute__((ext_vector_type(2))) float v2f;
typedef __attribute__((ext_vector_type(8))) float v8f;

// ---------------------------------------------------------------------------
// threefry2x32 (Random123 / jax.random) — exact, for the dropout mask.
// jax.random.key(42) -> key = (0, 42).
// ---------------------------------------------------------------------------
__device__ __forceinline__ unsigned tf_rotl(unsigned x, int r) {
    return (x << r) | (x >> (32 - r));
}

__device__ __forceinline__ void threefry2x32(unsigned k0, unsigned k1,
                                             unsigned x0, unsigned x1,
                                             unsigned& o0, unsigned& o1) {
    const unsigned ks[3] = {k0, k1, 0x1BD11BDAu ^ k0 ^ k1};
    const int R[8] = {13, 15, 26, 6, 17, 29, 16, 24};
    x0 += ks[0]; x1 += ks[1];
#pragma unroll
    for (int i = 0; i < 20; ++i) {
        x0 += x1;
        x1 = tf_rotl(x1, R[i & 7]);
        x1 ^= x0;
        if ((i & 3) == 3) {
            const unsigned j = (unsigned)(i >> 2) + 1u;   // 1..5
            x0 += ks[j % 3];
            x1 += ks[(j + 1u) % 3] + j;
        }
    }
    o0 = x0; o1 = x1;
}

// ---------------------------------------------------------------------------
// Utility kernels
// ---------------------------------------------------------------------------
__global__ void gcn_zero_f32(float* __restrict__ p, long long n) {
    long long i = (long long)blockIdx.x * blockDim.x + threadIdx.x;
    if (i < n) p[i] = 0.0f;
}

__global__ void gcn_init_out_bias(float* __restrict__ out, const float* __restrict__ b2) {
    long long i = (long long)blockIdx.x * blockDim.x + threadIdx.x;
    if (i < (long long)N_NODES * OUT_C) out[i] = b2[i & (OUT_C - 1)];
}

__global__ void gcn_degree(const int* __restrict__ dst, float* __restrict__ deg) {
    int e = blockIdx.x * blockDim.x + threadIdx.x;
    if (e < N_EDGES) atomicAdd(&deg[dst[e]], 1.0f);
}

__global__ void gcn_dinv(const float* __restrict__ deg, float* __restrict__ dinv) {
    int i = blockIdx.x * blockDim.x + threadIdx.x;
    // +1 for the self-loop; degree is therefore always > 0.
    if (i < N_NODES) dinv[i] = rsqrtf(deg[i] + 1.0f);
}

// ---------------------------------------------------------------------------
// WMMA GEMM: C[M x N] = A[M x 128] * W[128 x N], fp32 via V_WMMA_F32_16X16X4_F32.
// Block = 128 threads (4 waves), each wave owns a 16-row strip.
//
// W is staged in LDS *transposed* with padded stride 132 (lwT[col*132 + k]) so
// that each B fragment {W[k][col], W[k+1][col]} is one aligned ds_load_b64
// into an even VGPR pair, and the 16 column-lanes hit disjoint bank pairs
// (132 mod 64 = 4).
//
// A-row tail handling: row index clamped to M-1 (always-in-bounds load, no
// EXEC divergence in the MMA loop). A row M only contributes to C row M, and
// out-of-range C rows are never stored, so the clamp is safe.
//
// Operand layout (ISA 7.12.2, 32-bit, wave32):
//   A 16x4 : lanes 0-15 carry M=lane  {v0:K=0, v1:K=1}; lanes 16-31 {K=2,K=3}
//   B 4x16 : lanes 0-15 carry N=lane  {v0:K=0, v1:K=1}; lanes 16-31 {K=2,K=3}
//   C 16x16: VGPR i -> M=i (lanes 0-15) / M=i+8 (lanes 16-31), N = lane&15
// ---------------------------------------------------------------------------
#define LW_STRIDE 132

template <int N>
__global__ __launch_bounds__(128)
void gcn_gemm_wmma(const float* __restrict__ A, const float* __restrict__ W,
                   float* __restrict__ C, int M) {
    __shared__ float lwT[N * LW_STRIDE];

    const int tid  = threadIdx.x;
    const int lane = tid & 31;
    const int wave = tid >> 5;

    // cooperative transposed panel load: read W row-major (coalesced float4),
    // write lwT[col*132 + k]
    {
        const float4* Wv = (const float4*)W;
        const int nv4 = (128 * N) / 4;
        for (int i = tid; i < nv4; i += 128) {
            const float4 w = Wv[i];
            const int k  = (i * 4) / N;
            const int n0 = (i * 4) % N;
            lwT[(n0 + 0) * LW_STRIDE + k] = w.x;
            lwT[(n0 + 1) * LW_STRIDE + k] = w.y;
            lwT[(n0 + 2) * LW_STRIDE + k] = w.z;
            lwT[(n0 + 3) * LW_STRIDE + k] = w.w;
        }
    }
    __syncthreads();

    const int rowBase = blockIdx.x * 64 + wave * 16;
    const int half    = lane >> 4;    // 0: K=0,1   1: K=2,3
    const int l16     = lane & 15;
    const int aRow    = min(rowBase + l16, M - 1);     // clamp: no divergence
    const float* aPtr = A + (size_t)aRow * 128 + half * 2;

    constexpr int NT = N / 16;
    const float* bPtr[NT];
#pragma unroll
    for (int t = 0; t < NT; ++t)
        bPtr[t] = lwT + (t * 16 + l16) * LW_STRIDE + half * 2;

    v8f acc[NT];
#pragma unroll
    for (int t = 0; t < NT; ++t) acc[t] = (v8f){0.f, 0.f, 0.f, 0.f, 0.f, 0.f, 0.f, 0.f};

    for (int k0 = 0; k0 < 128; k0 += 4) {
        const v2f aF = *(const v2f*)(aPtr + k0);       // 8B-aligned b64 load
#pragma unroll
        for (int t = 0; t < NT; ++t) {
            const v2f bF = *(const v2f*)(bPtr[t] + k0); // single ds_load_b64
            acc[t] = __builtin_amdgcn_wmma_f32_16x16x4_f32(
                /*neg_a=*/false, aF, /*neg_b=*/false, bF,
                /*c_mod=*/(short)0, acc[t], /*reuse_a=*/false, /*reuse_b=*/false);
        }
    }

#pragma unroll
    for (int t = 0; t < NT; ++t) {
#pragma unroll
        for (int i = 0; i < 8; ++i) {
            const int r = rowBase + i + half * 8;
            if (r < M) C[(size_t)r * N + t * 16 + l16] = acc[t][i];
        }
    }
}

// ---------------------------------------------------------------------------
// Message passing: out[dst] += H[src] * dinv[src]*dinv[dst] over E real edges
// plus N self-loops (e in [E, E+N) -> src=dst=e-E). 4 channels per thread.
// ---------------------------------------------------------------------------
template <int C>
__global__ void gcn_scatter(const float* __restrict__ H,
                            const int* __restrict__ src, const int* __restrict__ dst,
                            const float* __restrict__ dinv, float* __restrict__ out) {
    constexpr int C4 = C / 4;
    const long long total = (long long)(N_EDGES + N_NODES) * C4;
    long long gid = (long long)blockIdx.x * blockDim.x + threadIdx.x;
    if (gid >= total) return;

    const int e = (int)(gid / C4);
    const int c = (int)(gid % C4) * 4;

    int s, d;
    if (e < N_EDGES) { s = src[e]; d = dst[e]; }
    else             { s = d = e - N_EDGES; }

    const float nrm = dinv[s] * dinv[d];
    const float4 hv = *(const float4*)(H + (size_t)s * C + c);
    float* o = out + (size_t)d * C + c;
    atomicAdd(o + 0, hv.x * nrm);
    atomicAdd(o + 1, hv.y * nrm);
    atomicAdd(o + 2, hv.z * nrm);
    atomicAdd(o + 3, hv.w * nrm);
}

// ---------------------------------------------------------------------------
// h = dropout(relu(agg + b1)), p=0.5, jax threefry key (0,42), scale x2.
// jax random_bits layout: counts = iota(NC), split into halves; element i<half
// takes out0 of threefry(i, i+half), else out1 of threefry(i-half, i).
// uniform u = bitcast((bits>>9)|0x3f800000) - 1 ; keep = u < 0.5.
// ---------------------------------------------------------------------------
__global__ void gcn_bias_relu_dropout(const float* __restrict__ agg,
                                      const float* __restrict__ b1,
                                      float* __restrict__ h) {
    const long long n = (long long)N_NODES * HID_C;
    long long i = (long long)blockIdx.x * blockDim.x + threadIdx.x;
    if (i >= n) return;

    float v = agg[i] + b1[i & (HID_C - 1)];
    v = fmaxf(v, 0.0f);

    const unsigned halfn = (unsigned)(n / 2);
    const bool lo = (i < (long long)halfn);
    const unsigned x0 = lo ? (unsigned)i : (unsigned)i - halfn;
    const unsigned x1 = lo ? (unsigned)i + halfn : (unsigned)i;
    unsigned o0, o1;
    threefry2x32(0u, 42u, x0, x1, o0, o1);
    const unsigned bits = lo ? o0 : o1;
    const float u = __uint_as_float((bits >> 9) | 0x3f800000u) - 1.0f;

    h[i] = (u < 0.5f) ? v * 2.0f : 0.0f;
}

// ---------------------------------------------------------------------------
// Driver
// ---------------------------------------------------------------------------
static inline int nblk(long long n, int b) { return (int)((n + b - 1) / b); }

extern "C" void kernel_launch(void* const* d_in, const int* in_sizes, int n_in,
                              void* d_out, int out_size, void* d_ws, size_t ws_size,
                              hipStream_t stream) {
    const float* x  = (const float*)d_in[0];   // [50000,128]
    const float* W1 = (const float*)d_in[1];   // [128,128]
    const float* b1 = (const float*)d_in[2];   // [128]
    const float* W2 = (const float*)d_in[3];   // [128,64]
    const float* b2 = (const float*)d_in[4];   // [64]
    const int*   ei = (const int*)d_in[5];     // [2,600000]
    const int* src = ei;
    const int* dst = ei + N_EDGES;

    float* out = (float*)d_out;                // [50000,64]

    // workspace carve-up (~64.4 MB)
    float* deg  = (float*)d_ws;                                  // N
    float* dinv = deg  + N_NODES;                                // N
    float* h1   = dinv + N_NODES;                                // N*128 (reused post-dropout)
    float* agg1 = h1   + (size_t)N_NODES * HID_C;                // N*128
    float* h2   = agg1 + (size_t)N_NODES * HID_C;                // N*64

    const int B = 256;

    // 1) init: deg=0, agg1=0, out=b2 broadcast
    gcn_zero_f32<<<nblk(N_NODES, B), B, 0, stream>>>(deg, N_NODES);
    gcn_zero_f32<<<nblk((long long)N_NODES * HID_C, B), B, 0, stream>>>(
        agg1, (long long)N_NODES * HID_C);
    gcn_init_out_bias<<<nblk((long long)N_NODES * OUT_C, B), B, 0, stream>>>(out, b2);

    // 2) degree + 3) rsqrt norm
    gcn_degree<<<nblk(N_EDGES, B), B, 0, stream>>>(dst, deg);
    gcn_dinv<<<nblk(N_NODES, B), B, 0, stream>>>(deg, dinv);

    // 4) H1 = X @ W1 (WMMA fp32)
    gcn_gemm_wmma<HID_C><<<nblk(N_NODES, 64), 128, 0, stream>>>(x, W1, h1, N_NODES);

    // 5) agg1[d] += H1[s] * norm
    gcn_scatter<HID_C><<<nblk((long long)(N_EDGES + N_NODES) * (HID_C / 4), B), B, 0, stream>>>(
        h1, src, dst, dinv, agg1);

    // 6) h = dropout(relu(agg1 + b1))  (overwrites h1)
    gcn_bias_relu_dropout<<<nblk((long long)N_NODES * HID_C, B), B, 0, stream>>>(agg1, b1, h1);

    // 7) H2 = h @ W2 (WMMA fp32)
    gcn_gemm_wmma<OUT_C><<<nblk(N_NODES, 64), 128, 0, stream>>>(h1, W2, h2, N_NODES);

    // 8) out[d] += H2[s] * norm  (out pre-initialized with b2)
    gcn_scatter<OUT_C><<<nblk((long long)(N_EDGES + N_NODES) * (OUT_C / 4), B), B, 0, stream>>>(
        h2, src, dst, dinv, out);
}